// CrossNetwork_6322191860060
// MI455X (gfx1250) — compile-verified
//
#include <hip/hip_runtime.h>

#define L      1024
#define BROWS  16384
#define NCHUNK 64
#define EPS    1e-8f

typedef float v2f __attribute__((ext_vector_type(2)));
typedef float v8f __attribute__((ext_vector_type(8)));

// ---------------------------------------------------------------------------
// Kernel 1: per-column partial sum / sum-of-squares via V_WMMA_F32_16X16X4_F32.
// A = ones(16x4)  =>  D[m,n] = sum_k B[k,n]  (column sums of a 4x16 tile),
// accumulated in the f32 C operand across the row loop. Result is identical in
// every row m of D, so element 0 of the accumulator holds colsum for n=lane&15.
// Grid: (L/16 column tiles, NCHUNK row chunks). Block: 256 = 8 waves.
// ---------------------------------------------------------------------------
__global__ __launch_bounds__(256)
void stats_partial(const float* __restrict__ xin,
                   float* __restrict__ psum, float* __restrict__ psq)
{
    const int lane = threadIdx.x & 31;
    const int wv   = threadIdx.x >> 5;      // wave 0..7
    const int n    = lane & 15;             // B/D column (N = lane % 16)
    const int kh   = lane >> 4;             // which pair of K rows this half-wave holds
    const int col  = blockIdx.x * 16 + n;

    const int rows_per_block = BROWS / NCHUNK;      // 256
    const int rows_per_wave  = rows_per_block / 8;  // 32
    const int r0 = blockIdx.y * rows_per_block + wv * rows_per_wave + kh * 2;

    v2f a; a[0] = 1.0f; a[1] = 1.0f;   // ones A-matrix: layout-independent
    v8f csum = {};
    v8f csq  = {};

    const float* p = xin + (size_t)r0 * L + col;
    #pragma unroll
    for (int it = 0; it < rows_per_wave / 4; ++it) {
        v2f b;
        b[0] = p[0];        // row r0 + it*4
        b[1] = p[L];        // row r0 + it*4 + 1
        v2f b2; b2[0] = b[0] * b[0]; b2[1] = b[1] * b[1];
        csum = __builtin_amdgcn_wmma_f32_16x16x4_f32(false, a, false, b,
                                                     (short)0, csum, false, false);
        csq  = __builtin_amdgcn_wmma_f32_16x16x4_f32(false, a, false, b2,
                                                     (short)0, csq,  false, false);
        p += 4 * L;
    }

    __shared__ float lsum[8][16];
    __shared__ float lsq [8][16];
    if (lane < 16) { lsum[wv][n] = csum[0]; lsq[wv][n] = csq[0]; }
    __syncthreads();
    if (threadIdx.x < 16) {
        float s1 = 0.f, s2 = 0.f;
        #pragma unroll
        for (int i = 0; i < 8; ++i) { s1 += lsum[i][threadIdx.x]; s2 += lsq[i][threadIdx.x]; }
        const int c = blockIdx.x * 16 + threadIdx.x;
        psum[blockIdx.y * L + c] = s1;
        psq [blockIdx.y * L + c] = s2;
    }
}

// ---------------------------------------------------------------------------
// Kernel 2: fold NCHUNK partials -> scale/shift/coef per column + scalar
// C = sum_l shift[l]*w[l]. Single workgroup of 1024 threads (one per column).
// ---------------------------------------------------------------------------
__global__ __launch_bounds__(1024)
void finalize_stats(const float* __restrict__ psum, const float* __restrict__ psq,
                    const float* __restrict__ gamma, const float* __restrict__ beta,
                    const float* __restrict__ w,
                    float* __restrict__ scale_o, float* __restrict__ shift_o,
                    float* __restrict__ coef_o,  float* __restrict__ C_o)
{
    const int l = threadIdx.x;
    float s1 = 0.f, s2 = 0.f;
    for (int p = 0; p < NCHUNK; ++p) { s1 += psum[p * L + l]; s2 += psq[p * L + l]; }

    const float inv_b = 1.0f / (float)BROWS;
    const float mu  = s1 * inv_b;
    float       var = fmaxf(s2 * inv_b - mu * mu, 0.0f);   // biased variance
    const float sc  = gamma[l] / sqrtf(var + EPS);
    const float sh  = beta[l] - mu * sc;
    const float wl  = w[l];
    scale_o[l] = sc;
    shift_o[l] = sh;
    coef_o [l] = sc * wl;

    // reduce C = sum_l sh*wl across 1024 threads (32 waves)
    float cp = sh * wl;
    for (int off = 16; off; off >>= 1) cp += __shfl_xor(cp, off, 32);
    __shared__ float wpart[32];
    if ((threadIdx.x & 31) == 0) wpart[threadIdx.x >> 5] = cp;
    __syncthreads();
    if (threadIdx.x == 0) {
        float c = 0.f;
        #pragma unroll
        for (int i = 0; i < 32; ++i) c += wpart[i];
        *C_o = c;
    }
}

// ---------------------------------------------------------------------------
// Kernel 3: per-row s = x . coef + C, then out = x0*s + bias + (x*scale+shift).
// 256 threads per row (4 cols/thread, fixed => per-thread params live in regs).
// NOTE: xin/out alias in layers 1-2 (in-place) and x0 aliases xin in layer 0,
// so these pointers are deliberately NOT __restrict__. In-place is safe: all
// reads of a row precede the barrier that precedes its writes, and each row is
// owned by exactly one block.
// ---------------------------------------------------------------------------
__global__ __launch_bounds__(256)
void row_update(const float* xin, const float* x0,
                const float* __restrict__ bias,
                const float* __restrict__ scale, const float* __restrict__ shift,
                const float* __restrict__ coef,  const float* __restrict__ Cp,
                float* out, int rows_per_block)
{
    const int t  = threadIdx.x;
    const int c4 = t * 4;
    const float4 cf = *(const float4*)(coef  + c4);
    const float4 sc = *(const float4*)(scale + c4);
    const float4 sh = *(const float4*)(shift + c4);
    const float4 bs = *(const float4*)(bias  + c4);
    const float  C  = *Cp;

    __shared__ float wsum[8];

    int r = blockIdx.x * rows_per_block;
    const int rend = r + rows_per_block;
    for (; r < rend; ++r) {
        const float* xr  = xin + (size_t)r * L + c4;
        const float* x0r = x0  + (size_t)r * L + c4;
        const float4 xv  = *(const float4*)xr;
        const float4 x0v = *(const float4*)x0r;
        if (r + 1 < rend) {                       // stream next row (global_prefetch_b8)
            __builtin_prefetch(xr  + L, 0, 3);
            __builtin_prefetch(x0r + L, 0, 3);
        }

        float pd = xv.x * cf.x + xv.y * cf.y + xv.z * cf.z + xv.w * cf.w;
        for (int off = 16; off; off >>= 1) pd += __shfl_xor(pd, off, 32);
        if ((t & 31) == 0) wsum[t >> 5] = pd;
        __syncthreads();

        float s = C;
        #pragma unroll
        for (int i = 0; i < 8; ++i) s += wsum[i];

        float4 o;
        o.x = fmaf(x0v.x, s, bs.x + fmaf(xv.x, sc.x, sh.x));
        o.y = fmaf(x0v.y, s, bs.y + fmaf(xv.y, sc.y, sh.y));
        o.z = fmaf(x0v.z, s, bs.z + fmaf(xv.z, sc.z, sh.z));
        o.w = fmaf(x0v.w, s, bs.w + fmaf(xv.w, sc.w, sh.w));
        *(float4*)(out + (size_t)r * L + c4) = o;
        __syncthreads();   // protect wsum reuse next iteration
    }
}

// ---------------------------------------------------------------------------
// Launcher: 3 layers. Layer 0: d_in.x -> d_out; layers 1,2: in-place on d_out.
// x0 is always the original input. Scratch: partials + per-column params.
// ---------------------------------------------------------------------------
extern "C" void kernel_launch(void* const* d_in, const int* in_sizes, int n_in,
                              void* d_out, int out_size, void* d_ws, size_t ws_size,
                              hipStream_t stream)
{
    const float* x     = (const float*)d_in[0];
    const float* gamma = (const float*)d_in[1];
    const float* beta  = (const float*)d_in[2];
    const float* w     = (const float*)d_in[3];
    const float* bvec  = (const float*)d_in[4];
    float* out = (float*)d_out;

    float* ws    = (float*)d_ws;
    float* psum  = ws;                    // [NCHUNK][L]
    float* psq   = psum + NCHUNK * L;     // [NCHUNK][L]
    float* scale = psq  + NCHUNK * L;     // [L]
    float* shift = scale + L;             // [L]
    float* coef  = shift + L;             // [L]
    float* Cp    = coef  + L;             // [1]

    const int RU_BLOCKS = 2048;
    const int RPB       = BROWS / RU_BLOCKS;

    const float* cur = x;
    for (int layer = 0; layer < 3; ++layer) {
        stats_partial<<<dim3(L / 16, NCHUNK), 256, 0, stream>>>(cur, psum, psq);
        finalize_stats<<<1, 1024, 0, stream>>>(psum, psq,
                                               gamma + layer * L, beta + layer * L,
                                               w + layer * L,
                                               scale, shift, coef, Cp);
        row_update<<<RU_BLOCKS, 256, 0, stream>>>(cur, x, bvec + layer * L,
                                                  scale, shift, coef, Cp,
                                                  out, RPB);
        cur = out;
    }
}